// MemoryInsDis_3083786519080
// MI455X (gfx1250) — compile-verified
//
#include <hip/hip_runtime.h>

// ---------------------------------------------------------------------------
// Problem constants (from the reference): B=256, D=128, N=1e6, K=4096, T=0.07
// ---------------------------------------------------------------------------
#define BB   256
#define DD   128
#define NN   1000000
#define KP1  4097          // K+1 columns of idx / out
#define KTILES 257         // ceil(4097/16)
#define T_INV (1.0f / 0.07f)
#define WAVES_PER_BLOCK 4

typedef __attribute__((ext_vector_type(2))) float v2f;
typedef __attribute__((ext_vector_type(8))) float v8f;

// ---------------------------------------------------------------------------
// Kernel 1: gathered scores via V_WMMA_F32_16X16X4_F32.
// One wave per 16-wide k-tile of one batch row b.
//   A (16x4 f32): lane m (0..15) / m+16 holds row idx[b, k0+m] at d-offsets
//                 {d0+2*pair, d0+2*pair+1}  (pair = lane>>4)  -> v2f
//   B (4x16 f32): all 16 N columns replicate x[b]; every lane supplies
//                 {x[d0+2*pair], x[d0+2*pair+1]}               -> v2f
//   C/D (16x16 f32, 8 VGPRs): column N=0 lives in lanes 0 (M=0..7) and
//                 16 (M=8..15); those two lanes store the 16 results.
// ---------------------------------------------------------------------------
__global__ void __launch_bounds__(32 * WAVES_PER_BLOCK)
insdis_scores(const float* __restrict__ x,
              const float* __restrict__ memory,
              const int*   __restrict__ idx,
              float*       __restrict__ out)
{
    __shared__ float xs[DD];

    const int b    = blockIdx.y;
    const int tid  = threadIdx.x;          // 0..127
    const int wave = tid >> 5;
    const int lane = tid & 31;

    // Stage x[b] (512 B) in LDS once per block.
    xs[tid] = x[b * DD + tid];
    __syncthreads();

    const int ktile = blockIdx.x * WAVES_PER_BLOCK + wave;   // wave-uniform
    if (ktile >= KTILES) return;                             // whole-wave exit

    const int m    = lane & 15;
    const int pair = lane >> 4;            // 0 or 1 (K half of the chunk)

    int k  = ktile * 16 + m;
    int kc = (k < KP1) ? k : (KP1 - 1);    // clamp tail for safe gather
    const long long row = (long long)idx[b * KP1 + kc] * DD;
    const float* __restrict__ arow = memory + row;

    v8f c = {};
    #pragma unroll
    for (int d0 = 0; d0 < DD; d0 += 4) {
        // A: 8-byte aligned lane-local piece of the gathered row
        v2f a  = *(const v2f*)(arow + d0 + 2 * pair);
        // B: same d-chunk of x[b] from LDS (identical across N columns)
        v2f bv = *(const v2f*)(xs + d0 + 2 * pair);
        c = __builtin_amdgcn_wmma_f32_16x16x4_f32(
                /*neg_a=*/false, a, /*neg_b=*/false, bv,
                /*c_mod=*/(short)0, c, /*reuse_a=*/false, /*reuse_b=*/false);
    }

    // Column N=0 of D: lane 0 holds M=0..7, lane 16 holds M=8..15.
    if (lane == 0 || lane == 16) {
        const int mbase = ktile * 16 + pair * 8;
        #pragma unroll
        for (int v = 0; v < 8; ++v) {
            const int kk = mbase + v;
            if (kk < KP1) out[b * KP1 + kk] = c[v] * T_INV;
        }
    }
}

// ---------------------------------------------------------------------------
// Kernel 2: momentum update + L2 normalize of the 256 positive rows,
// written over the already-copied new_memory. One block per b, 128 threads.
// ---------------------------------------------------------------------------
__global__ void __launch_bounds__(DD)
insdis_update(const float* __restrict__ x,
              const float* __restrict__ memory,
              const int*   __restrict__ y,
              float*       __restrict__ newmem)
{
    __shared__ float red[DD];

    const int b = blockIdx.x;
    const int d = threadIdx.x;
    const long long row = (long long)y[b] * DD;

    float v = 0.5f * memory[row + d] + 0.5f * x[b * DD + d];

    red[d] = v * v;
    __syncthreads();
    #pragma unroll
    for (int s = DD / 2; s > 0; s >>= 1) {
        if (d < s) red[d] += red[d + s];
        __syncthreads();
    }
    const float inv_norm = __frsqrt_rn(red[0]);
    newmem[row + d] = v * inv_norm;
}

// ---------------------------------------------------------------------------
// Launch: scores || (copy -> row update), all stream-ordered.
// d_in order: x [B*D] f32, memory [N*D] f32, y [B] i32, idx [B*(K+1)] i32
// d_out: out [B*(K+1)] f32  ++  new_memory [N*D] f32
// ---------------------------------------------------------------------------
extern "C" void kernel_launch(void* const* d_in, const int* in_sizes, int n_in,
                              void* d_out, int out_size, void* d_ws, size_t ws_size,
                              hipStream_t stream)
{
    const float* x      = (const float*)d_in[0];
    const float* memory = (const float*)d_in[1];
    const int*   y      = (const int*)d_in[2];
    const int*   idx    = (const int*)d_in[3];

    float* out    = (float*)d_out;                       // [256 * 4097]
    float* newmem = out + (size_t)BB * KP1;              // [1e6 * 128]

    // 1) Scores: 256 b-rows x 257 k-tiles, 4 waves/block.
    dim3 grid((KTILES + WAVES_PER_BLOCK - 1) / WAVES_PER_BLOCK, BB);
    insdis_scores<<<grid, 32 * WAVES_PER_BLOCK, 0, stream>>>(x, memory, idx, out);

    // 2) Bulk copy memory -> new_memory (512 MB, d2d, capture-safe).
    hipMemcpyAsync(newmem, memory, (size_t)NN * DD * sizeof(float),
                   hipMemcpyDeviceToDevice, stream);

    // 3) Overwrite the 256 positive rows (ordered after the copy).
    insdis_update<<<BB, DD, 0, stream>>>(x, memory, y, newmem);
}